// RotationInitializer_3212635537982
// MI455X (gfx1250) — compile-verified
//
#include <hip/hip_runtime.h>
#include <stdint.h>

typedef float v2f __attribute__((ext_vector_type(2)));
typedef float v8f __attribute__((ext_vector_type(8)));

#define HID    256
#define T_TOR  262144
#define NHEADS 4
#define NUPD   4
#define ROWS_H (T_TOR / NHEADS)   // 65536 rows per head
#define IN_DIM 1026
#define OUTD   12
#define LEAKYC 0.001f
#define LN_EPS 1e-5f
#define LDP    260                 // padded LDS row stride (floats) -> conflict-free

#define WMMA_F32(accv, av, bv) \
  __builtin_amdgcn_wmma_f32_16x16x4_f32(false, (av), false, (bv), (short)0, (accv), false, false)

__device__ __forceinline__ float leaky_act(float v) { return v > 0.f ? v : LEAKYC * v; }

// Async-copy 1KB (256 floats) global -> LDS, spread over the wave's 32 lanes
// (two b128 async loads per lane). Tracked by ASYNCcnt; caller must
// s_wait_asynccnt before reading the staged LDS data.
__device__ __forceinline__ void async_copy_1k(uint32_t lds_base, const float* src, int lane) {
  uint32_t l0 = lds_base + (uint32_t)lane * 32u;
  const float* g0 = src + lane * 8;
  asm volatile("global_load_async_to_lds_b128 %0, %1, off\n\t"
               "global_load_async_to_lds_b128 %2, %3, off"
               :: "v"(l0), "v"(g0), "v"(l0 + 16u), "v"(g0 + 4)
               : "memory");
}

__device__ __forceinline__ void wait_async0() {
  asm volatile("s_wait_asynccnt 0x0" ::: "memory");
}
__device__ __forceinline__ void wait_ds0() {
  asm volatile("s_wait_dscnt 0x0" ::: "memory");
}

// ---------------------------------------------------------------------------
// Kernel 1: dihedral angles -> (cos, sin) per torsion, stored in workspace
// ---------------------------------------------------------------------------
__global__ void dihedral_cs_kernel(const float* __restrict__ X,
                                   const int* __restrict__ tix,
                                   float2* __restrict__ cs)
{
  int t = blockIdx.x * blockDim.x + threadIdx.x;
  if (t >= T_TOR) return;
  int i0 = tix[4 * t + 0], i1 = tix[4 * t + 1];
  int i2 = tix[4 * t + 2], i3 = tix[4 * t + 3];

  float p0x = X[3 * i0], p0y = X[3 * i0 + 1], p0z = X[3 * i0 + 2];
  float p1x = X[3 * i1], p1y = X[3 * i1 + 1], p1z = X[3 * i1 + 2];
  float p2x = X[3 * i2], p2y = X[3 * i2 + 1], p2z = X[3 * i2 + 2];
  float p3x = X[3 * i3], p3y = X[3 * i3 + 1], p3z = X[3 * i3 + 2];

  float r1x = p1x - p0x, r1y = p1y - p0y, r1z = p1z - p0z;
  float r2x = p2x - p1x, r2y = p2y - p1y, r2z = p2z - p1z;
  float r3x = p3x - p2x, r3y = p3y - p2y, r3z = p3z - p2z;

  float n1x = r1y * r2z - r1z * r2y;
  float n1y = r1z * r2x - r1x * r2z;
  float n1z = r1x * r2y - r1y * r2x;
  float n2x = r2y * r3z - r2z * r3y;
  float n2y = r2z * r3x - r2x * r3z;
  float n2z = r2x * r3y - r2y * r3x;

  float r2n = sqrtf(r2x * r2x + r2y * r2y + r2z * r2z);
  float yv  = r2n * (r1x * n2x + r1y * n2y + r1z * n2z);
  float xv  = n1x * n2x + n1y * n2y + n1z * n2z;

  float h2 = xv * xv + yv * yv;
  float2 o;
  if (h2 > 0.f) {
    float inv = rsqrtf(h2);
    o.x = xv * inv;   // cos(atan2(y,x))
    o.y = yv * inv;   // sin(atan2(y,x))
  } else {
    o.x = 1.f; o.y = 0.f;
  }
  cs[t] = o;
}

// ---------------------------------------------------------------------------
// Kernel 2: per-head MLP pipeline. 64 threads = 2 waves; each wave owns a
// 16-row strip, full 256-wide hidden state in registers (v8f acc[16]).
// Epilogue vectors are async-staged into LDS behind each GEMM k-loop.
// ---------------------------------------------------------------------------
__global__ __launch_bounds__(64)
void torsion_mlp_kernel(const float* __restrict__ atom_emb,
                        const int*   __restrict__ tix,
                        const float2* __restrict__ cs,
                        const float* __restrict__ W0,
                        const float* __restrict__ b0,
                        const float* __restrict__ Wh,
                        const float* __restrict__ bh,
                        const float* __restrict__ g,
                        const float* __restrict__ beta,
                        const float* __restrict__ Wout,
                        const float* __restrict__ bout,
                        float*       __restrict__ out)
{
  __shared__ float buf[2][16][LDP];     // per-wave 16x256 hidden state (padded)
  __shared__ float stage[2][3 * HID];   // per-wave async-staged epilogue vectors

  const int head    = blockIdx.y;
  const int wv      = threadIdx.x >> 5;
  const int lane    = threadIdx.x & 31;
  const int half    = lane >> 4;          // 0: K0/K1 rows 0-7, 1: K2/K3 rows 8-15
  const int ln      = lane & 15;
  const int rowbase = blockIdx.x * 32 + wv * 16;   // row within this head

  const float* W0h   = W0   + (size_t)head * IN_DIM * HID;
  const float* b0h   = b0   + head * HID;
  const float* Wouth = Wout + (size_t)head * HID * OUTD;
  const float* bouth = bout + head * OUTD;

  const uint32_t sbase = (uint32_t)(uintptr_t)(&stage[wv][0]);  // LDS byte offset
  const float*   st    = &stage[wv][0];

  // stage layer-0 epilogue vectors: W0 rows 1024/1025 (cos/sin cols) + b0
  async_copy_1k(sbase,          W0h + (size_t)1024 * HID, lane);
  async_copy_1k(sbase + 1024u,  W0h + (size_t)1025 * HID, lane);
  async_copy_1k(sbase + 2048u,  b0h, lane);

  v8f acc[16];
  {
    v8f z = {};
#pragma unroll
    for (int nt = 0; nt < 16; ++nt) acc[nt] = z;
  }

  // ---- gather this lane's A-row atom indices (A-row = local row ln) ----
  const int tA = 4 * (rowbase + ln) + head;
  const float* emb0 = atom_emb + (size_t)tix[4 * tA + 0] * HID;
  const float* emb1 = atom_emb + (size_t)tix[4 * tA + 1] * HID;
  const float* emb2 = atom_emb + (size_t)tix[4 * tA + 2] * HID;
  const float* emb3 = atom_emb + (size_t)tix[4 * tA + 3] * HID;
  const float* embs[4] = {emb0, emb1, emb2, emb3};

  // ================= layer 0: feats(1026) @ W0 =================
#pragma unroll
  for (int a = 0; a < 4; ++a) {
    const float* ea   = embs[a];
    const float* wseg = W0h + (size_t)(a * HID) * HID;
    for (int kk = 0; kk < 64; ++kk) {
      const int k2 = kk * 4 + 2 * half;
      v2f av = *(const v2f*)(ea + k2);
#pragma unroll
      for (int nt = 0; nt < 16; ++nt) {
        const int col = nt * 16 + ln;
        v2f bv;
        bv.x = wseg[(size_t)k2 * HID + col];
        bv.y = wseg[(size_t)(k2 + 1) * HID + col];
        acc[nt] = WMMA_F32(acc[nt], av, bv);
      }
    }
  }

  // cos/sin rank-2 epilogue + bias + leaky, write h0 into LDS
  float2 csr[8];
#pragma unroll
  for (int r = 0; r < 8; ++r) {
    const int row = rowbase + r + 8 * half;
    csr[r] = cs[4 * row + head];
  }
  wait_async0();   // staged epilogue vectors now valid in LDS
#pragma unroll
  for (int nt = 0; nt < 16; ++nt) {
    const int col = nt * 16 + ln;
    const float wc = st[col];
    const float ws = st[HID + col];
    const float bb = st[2 * HID + col];
#pragma unroll
    for (int r = 0; r < 8; ++r) {
      float v = acc[nt][r] + csr[r].x * wc + csr[r].y * ws + bb;
      buf[wv][r + 8 * half][col] = leaky_act(v);
    }
  }

  // ================= hidden layers: GEMM + LayerNorm + leaky =================
  for (int l = 0; l < NUPD; ++l) {
    const float* Wl  = Wh   + (size_t)(head * NUPD + l) * HID * HID;
    const float* bl  = bh   + (head * NUPD + l) * HID;
    const float* gl  = g    + (head * NUPD + l) * HID;
    const float* bel = beta + (head * NUPD + l) * HID;
    __builtin_prefetch(Wl + (size_t)lane * 64, 0, 1);   // global_prefetch_b8

    // stage this layer's epilogue vectors; overlap with the GEMM below.
    // (drain pending DS reads of the previous stage contents first:
    //  async LDS writes are unordered w.r.t. DS ops)
    wait_ds0();
    async_copy_1k(sbase,          bl,  lane);
    async_copy_1k(sbase + 1024u,  gl,  lane);
    async_copy_1k(sbase + 2048u,  bel, lane);

    {
      v8f z = {};
#pragma unroll
      for (int nt = 0; nt < 16; ++nt) acc[nt] = z;
    }
    for (int kk = 0; kk < 64; ++kk) {
      const int k2 = kk * 4 + 2 * half;
      v2f av = *(const v2f*)(&buf[wv][ln][k2]);
#pragma unroll
      for (int nt = 0; nt < 16; ++nt) {
        const int col = nt * 16 + ln;
        v2f bv;
        bv.x = Wl[(size_t)k2 * HID + col];
        bv.y = Wl[(size_t)(k2 + 1) * HID + col];
        acc[nt] = WMMA_F32(acc[nt], av, bv);
      }
    }

    wait_async0();   // staged bias/gamma/beta now valid

    // bias + in-register LayerNorm statistics
    float sum[8], sq[8];
#pragma unroll
    for (int r = 0; r < 8; ++r) { sum[r] = 0.f; sq[r] = 0.f; }
#pragma unroll
    for (int nt = 0; nt < 16; ++nt) {
      const float bb = st[nt * 16 + ln];
#pragma unroll
      for (int r = 0; r < 8; ++r) {
        float v = acc[nt][r] + bb;
        acc[nt][r] = v;
        sum[r] += v;
        sq[r]  += v * v;
      }
    }
    // reduce across the 16 lanes sharing each row (xor 1,2,4,8 stays in-half)
#pragma unroll
    for (int off = 1; off < 16; off <<= 1) {
#pragma unroll
      for (int r = 0; r < 8; ++r) {
        sum[r] += __shfl_xor(sum[r], off, 32);
        sq[r]  += __shfl_xor(sq[r],  off, 32);
      }
    }
    float mu[8], rs[8];
#pragma unroll
    for (int r = 0; r < 8; ++r) {
      mu[r] = sum[r] * (1.f / 256.f);
      float var = sq[r] * (1.f / 256.f) - mu[r] * mu[r];
      rs[r] = rsqrtf(var + LN_EPS);
    }
    // normalize * g + beta, leaky, write back to LDS for next layer
#pragma unroll
    for (int nt = 0; nt < 16; ++nt) {
      const int col = nt * 16 + ln;
      const float gg = st[HID + col];
      const float be = st[2 * HID + col];
#pragma unroll
      for (int r = 0; r < 8; ++r) {
        float v = (acc[nt][r] - mu[r]) * rs[r] * gg + be;
        buf[wv][r + 8 * half][col] = leaky_act(v);
      }
    }
  }

  // ================= output layer: h(256) @ Wout(256x12) =================
  v8f oacc = {};
  for (int kk = 0; kk < 64; ++kk) {
    const int k2 = kk * 4 + 2 * half;
    v2f av = *(const v2f*)(&buf[wv][ln][k2]);
    v2f bv;
    bv.x = (ln < OUTD) ? Wouth[(size_t)k2 * OUTD + ln] : 0.f;
    bv.y = (ln < OUTD) ? Wouth[(size_t)(k2 + 1) * OUTD + ln] : 0.f;
    oacc = WMMA_F32(oacc, av, bv);
  }
  if (ln < OUTD) {
    const float bo = bouth[ln];
#pragma unroll
    for (int r = 0; r < 8; ++r) {
      const int row = rowbase + r + 8 * half;
      out[((size_t)head * ROWS_H + row) * OUTD + ln] = oacc[r] + bo;
    }
  }
}

// ---------------------------------------------------------------------------
extern "C" void kernel_launch(void* const* d_in, const int* in_sizes, int n_in,
                              void* d_out, int out_size, void* d_ws, size_t ws_size,
                              hipStream_t stream)
{
  const float* atom_emb = (const float*)d_in[0];
  const float* X        = (const float*)d_in[1];
  const int*   tidx     = (const int*)  d_in[2];
  const float* W0       = (const float*)d_in[3];
  const float* b0       = (const float*)d_in[4];
  const float* Wh       = (const float*)d_in[5];
  const float* bh       = (const float*)d_in[6];
  const float* g        = (const float*)d_in[7];
  const float* beta     = (const float*)d_in[8];
  const float* Wout     = (const float*)d_in[9];
  const float* bout     = (const float*)d_in[10];
  float* out = (float*)d_out;

  float2* cs = (float2*)d_ws;   // T_TOR float2 = 2 MB of scratch

  dihedral_cs_kernel<<<T_TOR / 256, 256, 0, stream>>>(X, tidx, cs);

  dim3 grid(ROWS_H / 32, NHEADS);   // 2048 x 4 blocks, 32 rows each
  torsion_mlp_kernel<<<grid, 64, 0, stream>>>(atom_emb, tidx, cs,
                                              W0, b0, Wh, bh, g, beta,
                                              Wout, bout, out);
}